// Luong_Attention_33062658245299
// MI455X (gfx1250) — compile-verified
//
#include <hip/hip_runtime.h>
#include <math.h>

// Luong attention, fully fused-ish 3-kernel pipeline for gfx1250 (MI455X).
// fp32 WMMA (V_WMMA_F32_16X16X4_F32) used for all four GEMMs.

typedef __attribute__((ext_vector_type(2))) float v2f;
typedef __attribute__((ext_vector_type(4))) float v4f;
typedef __attribute__((ext_vector_type(8))) float v8f;

#define WMMA_F32(a, b, c) \
  __builtin_amdgcn_wmma_f32_16x16x4_f32(false, (a), false, (b), (short)0, (c), false, false)

// Problem sizes (fixed by the reference)
constexpr int Bn = 16, Sn = 2048, Tn = 1024, En = 256, Dn = 256;

// ---------------------------------------------------------------------------
// Kernel 1: enc_proj[b,s,:] = enc_states[b,s,:] @ W_enc   (M=B*S=32768, N=256, K=256)
// One wave computes a 16x64 output tile (4 accumulators), K-loop step 4.
// ---------------------------------------------------------------------------
__global__ __launch_bounds__(256) void k_proj(const float* __restrict__ A,
                                              const float* __restrict__ Wm,
                                              float* __restrict__ Cout) {
  const int lane = threadIdx.x & 31;
  const int wave = threadIdx.x >> 5;
  const int tile = blockIdx.x * 8 + wave;   // 8192 tiles = (32768/16) * (256/64)
  const int m0 = (tile >> 2) << 4;
  const int n0 = (tile & 3) << 6;
  const int hf = lane >> 4;                 // lane half (K split)
  const int l16 = lane & 15;

  v8f acc[4] = {};
  const float* arow = A + (size_t)(m0 + l16) * En;
  for (int k = 0; k < En; k += 4) {
    v2f a = *(const v2f*)(arow + k + 2 * hf);        // A[m][k..k+1] / [k+2..k+3]
#pragma unroll
    for (int j = 0; j < 4; ++j) {
      const float* bp = Wm + (size_t)(k + 2 * hf) * Dn + n0 + 16 * j + l16;
      v2f b;
      b.x = bp[0];     // B[k+2h][n]
      b.y = bp[Dn];    // B[k+2h+1][n]
      acc[j] = WMMA_F32(a, b, acc[j]);
    }
  }
#pragma unroll
  for (int j = 0; j < 4; ++j)
#pragma unroll
    for (int r = 0; r < 8; ++r)
      Cout[(size_t)(m0 + r + 8 * hf) * Dn + n0 + 16 * j + l16] = acc[j][r];
}

// ---------------------------------------------------------------------------
// Kernel 2: flash attention. One wave owns one (b, 16-row target tile).
//   scores(16x16 per chunk) = Q(16x256) @ K^T, online softmax over S,
//   O(16x256) += P @ V.  Q staged in LDS; P transposed C->A layout via LDS.
// 2 waves / block -> 34 KB static LDS.
// ---------------------------------------------------------------------------
__global__ __launch_bounds__(64) void k_attn(const float* __restrict__ dec,
                                             const float* __restrict__ encp,
                                             const float* __restrict__ enc,
                                             const unsigned char* __restrict__ mask,
                                             float* __restrict__ ctx) {
  __shared__ float lds[2 * (16 * 256 + 16 * 16)];
  const int lane = threadIdx.x & 31;
  const int wave = threadIdx.x >> 5;            // 0..1
  const int gtile = blockIdx.x * 2 + wave;      // 0..1023 = B * (T/16)
  const int b = gtile >> 6;
  const int t0 = (gtile & 63) << 4;
  float* qs = lds + wave * (16 * 256 + 256);    // 16x256 Q tile
  float* ps = qs + 16 * 256;                    // 16x16 P transpose scratch
  const int hf = lane >> 4;
  const int l16 = lane & 15;

  // Stage Q tile (dec rows t0..t0+15) into LDS, row-major 16x256.
  const float* qg = dec + ((size_t)b * Tn + t0) * Dn;
  for (int r = 0; r < 16; ++r) {
    v4f q0 = *(const v4f*)(qg + r * Dn + 4 * lane);
    v4f q1 = *(const v4f*)(qg + r * Dn + 128 + 4 * lane);
    *(v4f*)(qs + r * 256 + 4 * lane) = q0;
    *(v4f*)(qs + r * 256 + 128 + 4 * lane) = q1;
  }

  v8f O[16] = {};               // 16 column-tiles of the 16x256 context accumulator
  float mst[8], lst[8];         // per-row running max / sum (row = r + 8*hf)
#pragma unroll
  for (int r = 0; r < 8; ++r) { mst[r] = -1e30f; lst[r] = 0.f; }

  const unsigned char* mb = mask + (size_t)b * Sn;
  const float* kb = encp + (size_t)b * Sn * Dn;
  const float* vb = enc + (size_t)b * Sn * En;

  for (int s0 = 0; s0 < Sn; s0 += 16) {
    if (mb[s0] == 0) break;     // prefix-valid mask: rest of S is invalid

    // ---- scores tile: sc = Q @ K^T (16x16), K rows = enc_proj[b, s0+n, :]
    v8f sc = {};
    const float* kp = kb + (size_t)(s0 + l16) * Dn;   // this lane's n-column of K^T
    for (int k = 0; k < Dn; k += 4) {
      v2f a = *(const v2f*)(qs + l16 * 256 + k + 2 * hf);
      v2f bf = *(const v2f*)(kp + k + 2 * hf);        // contiguous in k
      sc = WMMA_F32(a, bf, sc);
    }

    // ---- mask + online softmax (per-row stats live at (vgpr r, lane half))
    const bool valid = (mb[s0 + l16] != 0);
    float alpha[8];
#pragma unroll
    for (int r = 0; r < 8; ++r) {
      float s = valid ? sc[r] : -1e30f;
      float mx = s;
      mx = fmaxf(mx, __shfl_xor(mx, 1));
      mx = fmaxf(mx, __shfl_xor(mx, 2));
      mx = fmaxf(mx, __shfl_xor(mx, 4));
      mx = fmaxf(mx, __shfl_xor(mx, 8));
      const float mnew = fmaxf(mst[r], mx);
      const float p = __expf(s - mnew);
      sc[r] = p;
      float sum = p;
      sum += __shfl_xor(sum, 1);
      sum += __shfl_xor(sum, 2);
      sum += __shfl_xor(sum, 4);
      sum += __shfl_xor(sum, 8);
      alpha[r] = __expf(mst[r] - mnew);
      lst[r] = alpha[r] * lst[r] + sum;
      mst[r] = mnew;
    }
#pragma unroll
    for (int t = 0; t < 16; ++t)
#pragma unroll
      for (int r = 0; r < 8; ++r) O[t][r] *= alpha[r];

    // ---- transpose P (C layout -> A layout) through per-wave LDS scratch
#pragma unroll
    for (int r = 0; r < 8; ++r) ps[(r + 8 * hf) * 16 + l16] = sc[r];
    // same-wave DS ops are in-order; no barrier needed (private region)

    // ---- O += P(16x16) @ V(16x256), V rows = enc_states[b, s0+k, :]
    const float* vrow = vb + (size_t)s0 * En;
#pragma unroll
    for (int kk = 0; kk < 16; kk += 4) {
      v2f a = *(const v2f*)(ps + l16 * 16 + kk + 2 * hf);
#pragma unroll
      for (int t = 0; t < 16; ++t) {
        const float* vp = vrow + (size_t)(kk + 2 * hf) * En + t * 16 + l16;
        v2f bf;
        bf.x = vp[0];
        bf.y = vp[En];
        O[t] = WMMA_F32(a, bf, O[t]);
      }
    }
  }

  // ---- normalize and write context
  float inv[8];
#pragma unroll
  for (int r = 0; r < 8; ++r) inv[r] = 1.0f / lst[r];  // lst >= 1 (s=0 always valid)
  float* cb = ctx + ((size_t)b * Tn + t0) * En;
#pragma unroll
  for (int t = 0; t < 16; ++t)
#pragma unroll
    for (int r = 0; r < 8; ++r)
      cb[(size_t)(r + 8 * hf) * En + t * 16 + l16] = O[t][r] * inv[r];
}

// ---------------------------------------------------------------------------
// Kernel 3: out = tanh([ctx | dec] @ W_fin)   (M=B*T=16384, N=256, K=512)
// ---------------------------------------------------------------------------
__global__ __launch_bounds__(256) void k_final(const float* __restrict__ ctx,
                                               const float* __restrict__ dec,
                                               const float* __restrict__ Wf,
                                               float* __restrict__ out) {
  const int lane = threadIdx.x & 31;
  const int wave = threadIdx.x >> 5;
  const int tile = blockIdx.x * 8 + wave;   // 4096 tiles = (16384/16) * (256/64)
  const int m0 = (tile >> 2) << 4;
  const int n0 = (tile & 3) << 6;
  const int hf = lane >> 4;
  const int l16 = lane & 15;

  v8f acc[4] = {};
  const float* a1 = ctx + (size_t)(m0 + l16) * En;
  const float* a2 = dec + (size_t)(m0 + l16) * Dn;
  for (int k = 0; k < En + Dn; k += 4) {
    const float* ar = (k < En) ? (a1 + k) : (a2 + (k - En));
    v2f a = *(const v2f*)(ar + 2 * hf);
#pragma unroll
    for (int j = 0; j < 4; ++j) {
      const float* bp = Wf + (size_t)(k + 2 * hf) * Dn + n0 + 16 * j + l16;
      v2f b;
      b.x = bp[0];
      b.y = bp[Dn];
      acc[j] = WMMA_F32(a, b, acc[j]);
    }
  }
#pragma unroll
  for (int j = 0; j < 4; ++j)
#pragma unroll
    for (int r = 0; r < 8; ++r)
      out[(size_t)(m0 + r + 8 * hf) * Dn + n0 + 16 * j + l16] = tanhf(acc[j][r]);
}

// ---------------------------------------------------------------------------
extern "C" void kernel_launch(void* const* d_in, const int* in_sizes, int n_in,
                              void* d_out, int out_size, void* d_ws, size_t ws_size,
                              hipStream_t stream) {
  (void)in_sizes; (void)n_in; (void)out_size; (void)ws_size;
  const float* enc = (const float*)d_in[0];                 // (B,S,E) f32
  const float* dec = (const float*)d_in[1];                 // (B,T,D) f32
  const unsigned char* mask = (const unsigned char*)d_in[2]; // (B,S) bool (1 byte)
  const float* W_enc = (const float*)d_in[3];               // (E,D) f32
  const float* W_fin = (const float*)d_in[4];               // (E+D,D) f32
  float* out = (float*)d_out;                               // (B,T,D) f32

  float* encp = (float*)d_ws;                               // (B,S,D)  33.5 MB
  float* ctx = encp + (size_t)Bn * Sn * Dn;                 // (B,T,E)  16.8 MB

  k_proj<<<1024, 256, 0, stream>>>(enc, W_enc, encp);
  k_attn<<<512, 64, 0, stream>>>(dec, encp, enc, mask, ctx);
  k_final<<<512, 256, 0, stream>>>(ctx, dec, W_fin, out);
}